// hierarchical_generalized_poisson_with_skipped_cycles_model_82617990906130
// MI455X (gfx1250) — compile-verified
//
#include <hip/hip_runtime.h>
#include <math.h>

// ---------------------------------------------------------------------------
// log_Z[i,m,s] = log(s1) - lam*s1 + logsumexp_x( a*x + g(s,x) )
//   a      = log(lam) + lam/41                  (per lambda)
//   g(s,x) = (x-1)*log(s1 - x/41) - lgamma(x+1) (shared across all lambdas)
// Valid support: x in [0, 41*s1 - 1]; the reference's eps-clamp never binds
// there (min term = lam/41 >= 0.012 >> 1e-32), so it is dropped.
// Memory traffic is negligible (512 B in / 51 KB out); the kernel is
// transcendental-VALU bound (~26.5M exp+fma). WMMA is inapplicable: the only
// GEMM factorization sum_x t^x * exp(g) overflows/underflows fp32 by
// thousands of nats and separable chunk scaling cannot fix it.
// ---------------------------------------------------------------------------

#define NTHREADS 512
#define SLICES   4
#define IM_COUNT 128    // 8 * 16 lambdas
#define S_MAX    100
#define X_TOTAL  4100   // 41 * S_MAX
#define X_PADDED 4112   // X_TOTAL rounded up to a multiple of 16
#define NEG_BIG  (-3.0e38f)

#ifndef __has_builtin
#define __has_builtin(x) 0
#endif

#if __has_builtin(__builtin_amdgcn_global_load_async_to_lds_b32)
#define HAVE_ASYNC_LDS 1
#define AS1 __attribute__((address_space(1)))
#define AS3 __attribute__((address_space(3)))
#else
#define HAVE_ASYNC_LDS 0
#endif

#if __has_builtin(__builtin_amdgcn_s_wait_asynccnt)
#define WAIT_ASYNCCNT0() __builtin_amdgcn_s_wait_asynccnt(0)
#else
#define WAIT_ASYNCCNT0() asm volatile("s_wait_asynccnt 0" ::: "memory")
#endif

// Kernel A: shared factorial table, lgt[x] = lgamma(x+1), x = 0..X_TOTAL-1
__global__ void lgamma_table_kernel(float* __restrict__ lgt, int n) {
  int x = blockIdx.x * blockDim.x + threadIdx.x;
  if (x < n) lgt[x] = lgammaf((float)x + 1.0f);
}

// Kernel B: one block per s; 512 threads = 4 x-slices x 128 lambdas.
__global__ __launch_bounds__(NTHREADS)
void gp_logz_kernel(const float* __restrict__ lambdas,  // [128]
                    const float* __restrict__ lgt,      // [X_TOTAL] or null
                    float* __restrict__ out) {          // [128, 100]
  __shared__ __align__(16) float g[X_PADDED];  // g(s,x) row for this block's s
  __shared__ float pm[NTHREADS];               // per-thread partial max
  __shared__ float ps[NTHREADS];               // per-thread partial sum

  const int   sidx = (int)blockIdx.x;      // 0..99
  const float s1f  = (float)(sidx + 1);
  const int   Xs   = 41 * (sidx + 1);      // series length for this s
  const int   Xsp  = (Xs + 15) & ~15;      // padded to vector-loop granule
  const int   tid  = (int)threadIdx.x;
  constexpr float inv41 = 1.0f / 41.0f;

  // --- Stage lgamma table tile into LDS (CDNA5 async global->LDS path) ---
  if (lgt != nullptr) {
#if HAVE_ASYNC_LDS
    for (int x = tid; x < Xs; x += NTHREADS) {
      __builtin_amdgcn_global_load_async_to_lds_b32(
          (AS1 int*)(void*)(lgt + x),
          (AS3 int*)(void*)(&g[x]),
          /*offset=*/0, /*cpol=*/0);
    }
    WAIT_ASYNCCNT0();     // wave's own async loads complete
    __syncthreads();      // all waves' LDS writes visible
#else
    for (int x = tid; x < Xs; x += NTHREADS) g[x] = lgt[x];
    __syncthreads();
#endif
  }

  // --- Build g(s,x) row in place; pad with NEG_BIG (exp -> 0, max no-op) ---
  for (int x = tid; x < Xs; x += NTHREADS) {
    float xf = (float)x;
    float lg = (lgt != nullptr) ? g[x] : lgammaf(xf + 1.0f);
    float u  = s1f - xf * inv41;               // > 0 on valid support
    g[x] = fmaf(xf - 1.0f, logf(u), -lg);
  }
  for (int x = Xs + tid; x < Xsp; x += NTHREADS) g[x] = NEG_BIG;
  __syncthreads();

  // --- Two-pass logsumexp; 4-contiguous x per thread -> ds_load_b128,
  //     lanes in a wave share x (LDS broadcast reads) ---
  const int   im    = tid & (IM_COUNT - 1);    // constant slice per wave32
  const int   slice = tid >> 7;                // 0..3
  const float lam   = lambdas[im];
  const float a     = fmaf(lam, inv41, logf(lam));
  const float4* g4  = reinterpret_cast<const float4*>(g);

  // Pass 1: max of v(x) = a*x + g(x).
  float m = NEG_BIG;
  for (int x0 = slice * 4; x0 < Xsp; x0 += 4 * SLICES) {
    float4 gv = g4[x0 >> 2];
    float  xf = (float)x0;
    float  v0 = fmaf(a, xf,        gv.x);
    float  v1 = fmaf(a, xf + 1.0f, gv.y);
    float  v2 = fmaf(a, xf + 2.0f, gv.z);
    float  v3 = fmaf(a, xf + 3.0f, gv.w);
    m = fmaxf(m, fmaxf(fmaxf(v0, v1), fmaxf(v2, v3)));
  }
  pm[tid] = m;
  __syncthreads();
  const float M = fmaxf(fmaxf(pm[im], pm[im + 128]),
                        fmaxf(pm[im + 256], pm[im + 384]));

  // Pass 2: sum of exp(v - M).
  float ssum = 0.0f;
  for (int x0 = slice * 4; x0 < Xsp; x0 += 4 * SLICES) {
    float4 gv = g4[x0 >> 2];
    float  xf = (float)x0;
    float  e0 = __expf(fmaf(a, xf,        gv.x) - M);
    float  e1 = __expf(fmaf(a, xf + 1.0f, gv.y) - M);
    float  e2 = __expf(fmaf(a, xf + 2.0f, gv.z) - M);
    float  e3 = __expf(fmaf(a, xf + 3.0f, gv.w) - M);
    ssum += (e0 + e1) + (e2 + e3);
  }
  ps[tid] = ssum;
  __syncthreads();

  if (slice == 0) {
    float tot = (ps[im] + ps[im + 128]) + (ps[im + 256] + ps[im + 384]);
    // log(lam) cancels between the prefactor and the series constant.
    out[im * S_MAX + sidx] = logf(s1f) - lam * s1f + M + logf(tot);
  }
}

extern "C" void kernel_launch(void* const* d_in, const int* in_sizes, int n_in,
                              void* d_out, int out_size, void* d_ws, size_t ws_size,
                              hipStream_t stream) {
  (void)in_sizes; (void)n_in; (void)out_size;
  const float* lambdas = (const float*)d_in[0];   // [8,16] f32
  // d_in[1] is s_max == 100 (static in the reference's trace shapes).
  float* out = (float*)d_out;                     // [8,16,100] f32

  float* lgt = nullptr;
  if (ws_size >= (size_t)X_TOTAL * sizeof(float)) {
    lgt = (float*)d_ws;
    lgamma_table_kernel<<<(X_TOTAL + 255) / 256, 256, 0, stream>>>(lgt, X_TOTAL);
  }
  gp_logz_kernel<<<S_MAX, NTHREADS, 0, stream>>>(lambdas, lgt, out);
}